// AttentionOnDetail_56341380989224
// MI455X (gfx1250) — compile-verified
//
#include <hip/hip_runtime.h>
#include <math.h>

typedef float v2f __attribute__((ext_vector_type(2)));
typedef float v8f __attribute__((ext_vector_type(8)));

#define RMS_EPS 1.1920929e-07f

// ---------------------------------------------------------------------------
// Kernel A: per-patch score logits.  logit = dot(xp, pw) * rsqrt(mean(xp^2)+eps)
// grid = B*Np = 4096 blocks, 256 threads. One streaming pass over x (33.5 MB).
// ---------------------------------------------------------------------------
__global__ void score_kernel(const float* __restrict__ x,
                             const float* __restrict__ pw,
                             float* __restrict__ logits) {
  const int p = blockIdx.x;
  const float* xp = x + (size_t)p * 2048;
  const int t = threadIdx.x;
  float xx = 0.f, xw = 0.f;
#pragma unroll
  for (int i = 0; i < 2; ++i) {
    int idx = i * 1024 + t * 4;           // fully coalesced float4
    float4 v = *(const float4*)(xp + idx);
    float4 w = *(const float4*)(pw + idx);
    xx += v.x * v.x + v.y * v.y + v.z * v.z + v.w * v.w;
    xw += v.x * w.x + v.y * w.y + v.z * w.z + v.w * w.w;
  }
#pragma unroll
  for (int m = 16; m >= 1; m >>= 1) {
    xx += __shfl_xor(xx, m, 32);
    xw += __shfl_xor(xw, m, 32);
  }
  __shared__ float sxx[8], sxw[8];
  if ((t & 31) == 0) { sxx[t >> 5] = xx; sxw[t >> 5] = xw; }
  __syncthreads();
  if (t == 0) {
    float X = 0.f, W = 0.f;
    for (int i = 0; i < 8; ++i) { X += sxx[i]; W += sxw[i]; }
    logits[p] = W * rsqrtf(X * (1.0f / 2048.0f) + RMS_EPS);
  }
}

// ---------------------------------------------------------------------------
// Kernel B: top-4 of 512 logits per batch (tie-break lower index), sorted asc.
// grid = B blocks, 32 threads (one wave).
// ---------------------------------------------------------------------------
__global__ void topk_kernel(const float* __restrict__ logits,
                            int* __restrict__ Isel) {
  const int b = blockIdx.x;
  const int lane = threadIdx.x;
  __shared__ float v[512];
  for (int i = lane; i < 512; i += 32) v[i] = logits[b * 512 + i];
  __syncthreads();
  int sel[4];
  for (int r = 0; r < 4; ++r) {
    float bv = -__builtin_huge_valf();
    int bi = 0x7fffffff;
    for (int i = lane; i < 512; i += 32) {
      float val = v[i];
      if (val > bv || (val == bv && i < bi)) { bv = val; bi = i; }
    }
#pragma unroll
    for (int m = 16; m >= 1; m >>= 1) {
      float ov = __shfl_xor(bv, m, 32);
      int oi = __shfl_xor(bi, m, 32);
      if (ov > bv || (ov == bv && oi < bi)) { bv = ov; bi = oi; }
    }
    sel[r] = bi;
    if (lane == 0) v[bi] = -__builtin_huge_valf();
    __syncthreads();
  }
  if (lane == 0) {
    int a[4] = {sel[0], sel[1], sel[2], sel[3]};
    for (int i = 0; i < 3; ++i)
      for (int j = 0; j < 3 - i; ++j)
        if (a[j] > a[j + 1]) { int tmp = a[j]; a[j] = a[j + 1]; a[j + 1] = tmp; }
    for (int i = 0; i < 4; ++i) Isel[b * 4 + i] = a[i];
  }
}

// ---------------------------------------------------------------------------
// Kernel C: QKVG GEMM for the 64 selected tokens per batch.
// out(64x4096) = Xg(64x128) @ W(4096x128)^T using V_WMMA_F32_16X16X4_F32.
// grid = (32 n-chunks, 8 batches), 256 threads (8 waves, 4 tiles each).
// Xg staged in LDS with row pad 132 (4-float bank stride -> conflict-free).
// ---------------------------------------------------------------------------
__global__ void qkvg_kernel(const float* __restrict__ x,
                            const float* __restrict__ W,
                            const int* __restrict__ Isel,
                            float* __restrict__ qkvg) {
  const int nchunk = blockIdx.x;
  const int b = blockIdx.y;
  const int nbase = nchunk * 128;
  const int tid = threadIdx.x;
  __shared__ __align__(16) float Xg[64 * 132];
  __shared__ int sI[4];
  if (tid < 4) sI[tid] = Isel[b * 4 + tid];
  __syncthreads();
  for (int u = tid; u < 64 * 32; u += 256) {
    int row = u >> 5, c4 = (u & 31) * 4;
    size_t srow = (size_t)b * 8192 + (size_t)sI[row >> 4] * 16 + (row & 15);
    float4 v = *(const float4*)(x + srow * 128 + c4);
    *(float4*)(&Xg[row * 132 + c4]) = v;
  }
  __syncthreads();
  const int wid = tid >> 5, lane = tid & 31;
  const int lrow = lane & 15;
  const bool hi = lane >= 16;
  float* dst = qkvg + (size_t)b * 64 * 4096;
#pragma unroll
  for (int it = 0; it < 4; ++it) {
    int tile = wid + it * 8;
    int m0 = (tile & 3) * 16;
    int n0 = (tile >> 2) * 16;
    v8f acc = {0.f, 0.f, 0.f, 0.f, 0.f, 0.f, 0.f, 0.f};
    const float* wrow = W + (size_t)(nbase + n0 + lrow) * 128 + (hi ? 2 : 0);
    const float* arow = &Xg[(m0 + lrow) * 132 + (hi ? 2 : 0)];
    for (int kk = 0; kk < 32; ++kk) {
      v2f a, bf;
      float2 av = *(const float2*)(arow + kk * 4);
      float2 wv = *(const float2*)(wrow + kk * 4);
      a.x = av.x; a.y = av.y;
      bf.x = wv.x; bf.y = wv.y;
      acc = __builtin_amdgcn_wmma_f32_16x16x4_f32(false, a, false, bf,
                                                  (short)0, acc, false, false);
    }
    int ncol = nbase + n0 + lrow;
#pragma unroll
    for (int vv = 0; vv < 8; ++vv) {
      int m = m0 + vv + (hi ? 8 : 0);
      dst[(size_t)m * 4096 + ncol] = acc[vv];
    }
  }
}

// ---------------------------------------------------------------------------
// Kernel D: build q/k/v (B,65,H,128) with sink row, RoPE + rmsnorm*tao (q,k),
// v raw.  One wave per (b,s,h); lane holds 4 channels; RoPE pair via shfl_xor16.
// ---------------------------------------------------------------------------
__device__ __forceinline__ void gather4(const float* __restrict__ qkvg,
                                        const float* __restrict__ sink,
                                        int b, int s, int h, int c0, int off,
                                        float val[4]) {
  if (s == 0) {
#pragma unroll
    for (int i = 0; i < 4; ++i) val[i] = sink[h * 128 + c0 + i];
  } else {
    int sp = s - 1;
    int tok = ((sp >> 4) << 4) + off + ((sp & 15) >> 2);
    int chunk = sp & 3;
    const float* src = qkvg + (size_t)(b * 64 + tok) * 4096 + chunk * 1024 + h * 128 + c0;
    float4 v4 = *(const float4*)src;
    val[0] = v4.x; val[1] = v4.y; val[2] = v4.z; val[3] = v4.w;
  }
}

__device__ __forceinline__ void rope_norm_store(const float val[4],
                                                const float cv[4],
                                                const float sv[4],
                                                int lane, float tau,
                                                float* __restrict__ dst) {
  float part[4], o[4];
#pragma unroll
  for (int i = 0; i < 4; ++i) part[i] = __shfl_xor(val[i], 16, 32);
  if (lane < 16) {
#pragma unroll
    for (int i = 0; i < 4; ++i) o[i] = val[i] * cv[i] + part[i] * sv[i];
  } else {
#pragma unroll
    for (int i = 0; i < 4; ++i) o[i] = val[i] * cv[i] - part[i] * sv[i];
  }
  float ss = o[0] * o[0] + o[1] * o[1] + o[2] * o[2] + o[3] * o[3];
#pragma unroll
  for (int m = 16; m >= 1; m >>= 1) ss += __shfl_xor(ss, m, 32);
  float sc = rsqrtf(ss * (1.0f / 128.0f) + RMS_EPS) * tau;
#pragma unroll
  for (int i = 0; i < 4; ++i) dst[i] = o[i] * sc;
}

__global__ void prep_kernel(const float* __restrict__ qkvg,
                            const float* __restrict__ sink,
                            const float* __restrict__ cosp,
                            const float* __restrict__ sinp,
                            const float* __restrict__ tao,
                            float* __restrict__ qpost,
                            float* __restrict__ kpost,
                            float* __restrict__ vpost) {
  const int gw = blockIdx.x * 8 + (threadIdx.x >> 5);   // 0..4159
  const int lane = threadIdx.x & 31;
  const int b = gw / 520;
  const int rem = gw - b * 520;
  const int s = rem >> 3;
  const int h = rem & 7;
  const int c0 = lane * 4;
  const int cc = (lane & 15) * 4;
  float cv[4], sv[4];
#pragma unroll
  for (int i = 0; i < 4; ++i) {
    cv[i] = cosp[s * 64 + cc + i];
    sv[i] = sinp[s * 64 + cc + i];
  }
  const float t0 = tao[0], t1 = tao[1];
  const size_t obase = ((size_t)(b * 65 + s) * 8 + h) * 128 + c0;

  float qv[4], kv[4], vv[4];
  gather4(qkvg, sink, b, s, h, c0, 0, qv);
  gather4(qkvg, sink, b, s, h, c0, 4, kv);
  gather4(qkvg, sink, b, s, h, c0, 8, vv);
#pragma unroll
  for (int i = 0; i < 4; ++i) vpost[obase + i] = vv[i];   // v: raw
  rope_norm_store(qv, cv, sv, lane, t0, qpost + obase);
  rope_norm_store(kv, cv, sv, lane, t1, kpost + obase);
}

// ---------------------------------------------------------------------------
// Kernel E: causal softmax attention per (b,h), S=65, fused sigmoid gate.
// K rows in LDS (pad 129 -> lanes parallel over t hit distinct banks),
// V streamed coalesced from global (L2-resident).
// ---------------------------------------------------------------------------
__global__ void attn_kernel(const float* __restrict__ qpost,
                            const float* __restrict__ kpost,
                            const float* __restrict__ vpost,
                            const float* __restrict__ qkvg,
                            float* __restrict__ ybuf) {
  const int b = blockIdx.x >> 3;
  const int h = blockIdx.x & 7;
  const int tid = threadIdx.x;
  const int wid = tid >> 5, lane = tid & 31;
  __shared__ float kl[65 * 129];
  __shared__ float qrow[8][132];
  __shared__ float att[8][68];
  for (int u = tid; u < 65 * 32; u += 256) {
    int s = u >> 5, c4 = (u & 31) * 4;
    const float* src = kpost + ((size_t)(b * 65 + s) * 8 + h) * 128 + c4;
    float4 v = *(const float4*)src;
    kl[s * 129 + c4 + 0] = v.x;
    kl[s * 129 + c4 + 1] = v.y;
    kl[s * 129 + c4 + 2] = v.z;
    kl[s * 129 + c4 + 3] = v.w;
  }
  __syncthreads();
  const float inv = 0.08838834764831845f;  // 1/sqrt(128)
  const int c0 = lane * 4;
  for (int ri = 0; ri < 8; ++ri) {
    const int s = 1 + wid + ri * 8;        // rows 1..64 (row 0 is dropped)
    {
      const float* src = qpost + ((size_t)(b * 65 + s) * 8 + h) * 128 + c0;
#pragma unroll
      for (int i = 0; i < 4; ++i) qrow[wid][c0 + i] = src[i];
    }
    __syncthreads();
    float lmax = -__builtin_huge_valf();
    for (int t = lane; t <= s; t += 32) {
      float d = 0.f;
      const float* kr = &kl[t * 129];
      for (int c = 0; c < 128; ++c) d += qrow[wid][c] * kr[c];
      d *= inv;
      att[wid][t] = d;
      lmax = fmaxf(lmax, d);
    }
#pragma unroll
    for (int m = 16; m >= 1; m >>= 1) lmax = fmaxf(lmax, __shfl_xor(lmax, m, 32));
    float lsum = 0.f;
    for (int t = lane; t <= s; t += 32) {
      float e = expf(att[wid][t] - lmax);
      att[wid][t] = e;
      lsum += e;
    }
#pragma unroll
    for (int m = 16; m >= 1; m >>= 1) lsum += __shfl_xor(lsum, m, 32);
    __syncthreads();
    float acc[4] = {0.f, 0.f, 0.f, 0.f};
    for (int t = 0; t <= s; ++t) {
      float p = att[wid][t];
      const float* vr = vpost + ((size_t)(b * 65 + t) * 8 + h) * 128 + c0;
      float4 vv = *(const float4*)vr;
      acc[0] += p * vv.x; acc[1] += p * vv.y;
      acc[2] += p * vv.z; acc[3] += p * vv.w;
    }
    const float isum = 1.f / lsum;
    const int sp = s - 1;
    const int tok = ((sp >> 4) << 4) + 12 + ((sp & 15) >> 2);
    const int chunk = sp & 3;
    const float* gsrc = qkvg + (size_t)(b * 64 + tok) * 4096 + chunk * 1024 + h * 128 + c0;
    float4 g4 = *(const float4*)gsrc;
    float gv[4] = {g4.x, g4.y, g4.z, g4.w};
    float* dst = ybuf + (size_t)(b * 64 + sp) * 1024 + h * 128 + c0;
#pragma unroll
    for (int i = 0; i < 4; ++i) {
      float y = acc[i] * isum;
      dst[i] = y / (1.f + expf(-gv[i]));
    }
    __syncthreads();
  }
}

// ---------------------------------------------------------------------------
// Kernel F: output projection per batch: (64x1024) @ Wout(128x1024)^T -> 64x128
// f32 WMMA 16x16x4, K chunked by 128 with Y staged in LDS.
// ---------------------------------------------------------------------------
__global__ void proj_kernel(const float* __restrict__ ybuf,
                            const float* __restrict__ Wout,
                            float* __restrict__ out) {
  const int b = blockIdx.x;
  const int tid = threadIdx.x;
  const int wid = tid >> 5, lane = tid & 31;
  const int lrow = lane & 15;
  const bool hi = lane >= 16;
  __shared__ __align__(16) float Yl[64 * 132];
  v8f acc[4];
#pragma unroll
  for (int it = 0; it < 4; ++it) acc[it] = (v8f){0.f, 0.f, 0.f, 0.f, 0.f, 0.f, 0.f, 0.f};
  for (int kc = 0; kc < 8; ++kc) {
    __syncthreads();
    for (int u = tid; u < 64 * 32; u += 256) {
      int row = u >> 5, c4 = (u & 31) * 4;
      float4 v = *(const float4*)(ybuf + (size_t)(b * 64 + row) * 1024 + kc * 128 + c4);
      *(float4*)(&Yl[row * 132 + c4]) = v;
    }
    __syncthreads();
#pragma unroll
    for (int it = 0; it < 4; ++it) {
      int tile = wid + it * 8;
      int m0 = (tile & 3) * 16;
      int n0 = (tile >> 2) * 16;
      const float* arow = &Yl[(m0 + lrow) * 132 + (hi ? 2 : 0)];
      const float* wrow = Wout + (size_t)(n0 + lrow) * 1024 + kc * 128 + (hi ? 2 : 0);
      v8f a8 = acc[it];
      for (int kk = 0; kk < 32; ++kk) {
        v2f a, bf;
        float2 av = *(const float2*)(arow + kk * 4);
        float2 wv = *(const float2*)(wrow + kk * 4);
        a.x = av.x; a.y = av.y;
        bf.x = wv.x; bf.y = wv.y;
        a8 = __builtin_amdgcn_wmma_f32_16x16x4_f32(false, a, false, bf,
                                                   (short)0, a8, false, false);
      }
      acc[it] = a8;
    }
  }
#pragma unroll
  for (int it = 0; it < 4; ++it) {
    int tile = wid + it * 8;
    int m0 = (tile & 3) * 16;
    int n0 = (tile >> 2) * 16;
#pragma unroll
    for (int vv = 0; vv < 8; ++vv) {
      int m = m0 + vv + (hi ? 8 : 0);
      out[(size_t)(b * 64 + m) * 128 + n0 + lrow] = acc[it][vv];
    }
  }
}

// ---------------------------------------------------------------------------
// Host launcher
// ---------------------------------------------------------------------------
extern "C" void kernel_launch(void* const* d_in, const int* in_sizes, int n_in,
                              void* d_out, int out_size, void* d_ws, size_t ws_size,
                              hipStream_t stream) {
  const float* x       = (const float*)d_in[0];
  const float* cosp    = (const float*)d_in[1];
  const float* sinp    = (const float*)d_in[2];
  const float* sink    = (const float*)d_in[3];
  const float* Wqkvg   = (const float*)d_in[4];
  const float* patch_w = (const float*)d_in[5];
  const float* Wout    = (const float*)d_in[6];
  const float* tao     = (const float*)d_in[7];

  float* ws = (float*)d_ws;
  float* logits = ws;                       // 4096 floats
  int*   Isel   = (int*)(ws + 4096);        // 32 ints
  float* qkvg   = ws + 8192;                // 8*64*4096   = 2097152
  float* qpost  = qkvg + 2097152;           // 8*65*8*128  =  532480
  float* kpost  = qpost + 532480;
  float* vpost  = kpost + 532480;
  float* ybuf   = vpost + 532480;           // 8*64*1024   =  524288

  score_kernel<<<4096, 256, 0, stream>>>(x, patch_w, logits);
  topk_kernel<<<8, 32, 0, stream>>>(logits, Isel);
  qkvg_kernel<<<dim3(32, 8), 256, 0, stream>>>(x, Wqkvg, Isel, qkvg);
  prep_kernel<<<520, 256, 0, stream>>>(qkvg, sink, cosp, sinp, tao,
                                       qpost, kpost, vpost);
  attn_kernel<<<64, 256, 0, stream>>>(qpost, kpost, vpost, qkvg, ybuf);
  proj_kernel<<<8, 256, 0, stream>>>(ybuf, Wout, (float*)d_out);
}